// GCNLayer_36292473651564
// MI455X (gfx1250) — compile-verified
//
#include <hip/hip_runtime.h>
#include <stdint.h>

#define NNODES 50000
#define NEDGES 800000
#define DIM    128
#define LDSTR  132   // padded LDS row stride (dwords): banks 4*col+k, conflict-free
#define MCHUNK 5     // M-tiles per block in the GEMM (3125 = 625 * 5)

typedef __attribute__((ext_vector_type(2))) float        v2f;
typedef __attribute__((ext_vector_type(4))) float        v4f;
typedef __attribute__((ext_vector_type(8))) float        v8f;
typedef __attribute__((ext_vector_type(4))) unsigned int v4u;
typedef __attribute__((ext_vector_type(4))) int          v4i;
typedef __attribute__((ext_vector_type(8))) int          v8i;

// ---------------------------------------------------------------------------
// Zero an int/float array (n4 = count/4), B128 stores.
// ---------------------------------------------------------------------------
__global__ void zero_v4(float* __restrict__ p, int n4) {
    int i = blockIdx.x * blockDim.x + threadIdx.x;
    if (i < n4) {
        v4f z = {0.f, 0.f, 0.f, 0.f};
        *(v4f*)(p + (size_t)i * 4) = z;
    }
}

// ---------------------------------------------------------------------------
// CSR build step 1: histogram of dst (int atomics, randomly spread over N).
// ---------------------------------------------------------------------------
__global__ void count_edges(const long long* __restrict__ dst,
                            int* __restrict__ counts) {
    int e = blockIdx.x * blockDim.x + threadIdx.x;
    if (e < NEDGES) atomicAdd(counts + (int)dst[e], 1);
}

// ---------------------------------------------------------------------------
// CSR build step 2: exclusive scan of counts -> offsets[N+1], copy -> cursors.
// Single 1024-thread block, Hillis-Steele in LDS, 49 chunks. Negligible cost.
// ---------------------------------------------------------------------------
__global__ __launch_bounds__(1024) void scan_offsets(
        const int* __restrict__ counts,
        int* __restrict__ offsets,
        int* __restrict__ cursors, int n) {
    __shared__ int tmp[1024];
    __shared__ int carry_s;
    const int tid = threadIdx.x;
    if (tid == 0) carry_s = 0;
    __syncthreads();
    for (int base = 0; base < n; base += 1024) {
        const int i = base + tid;
        const int x = (i < n) ? counts[i] : 0;
        tmp[tid] = x;
        __syncthreads();
        for (int off = 1; off < 1024; off <<= 1) {
            int v = (tid >= off) ? tmp[tid - off] : 0;
            __syncthreads();
            tmp[tid] += v;
            __syncthreads();
        }
        const int excl = carry_s + tmp[tid] - x;
        if (i < n) { offsets[i] = excl; cursors[i] = excl; }
        __syncthreads();                 // everyone read carry_s
        if (tid == 1023) carry_s += tmp[1023];
        __syncthreads();
    }
    if (tid == 0) offsets[n] = carry_s;  // == NEDGES
}

// ---------------------------------------------------------------------------
// CSR build step 3: bucket fill. Returning int atomic gives slot; payload is
// one packed int2 {src, bits(w)} so the hot loop does a single b64 per edge.
// ---------------------------------------------------------------------------
__global__ void fill_edges(const long long* __restrict__ src,
                           const long long* __restrict__ dst,
                           const float*     __restrict__ w,
                           int*  __restrict__ cursors,
                           int2* __restrict__ epay) {
    int e = blockIdx.x * blockDim.x + threadIdx.x;
    if (e < NEDGES) {
        const int d   = (int)dst[e];
        const int pos = atomicAdd(cursors + d, 1);
        int2 p;
        p.x = (int)src[e];
        p.y = __float_as_int(w[e]);
        epay[pos] = p;
    }
}

// ---------------------------------------------------------------------------
// Aggregation: one wave32 per node. Lane L owns floats [4L,4L+4) of the row.
// Gathers feature[src] (L2-resident, global_load_b128), FMA-accumulates in
// registers, one plain B128 store per lane. No f32 atomics at all.
// Writes agg directly into d_out (GEMM then runs in place).
// ---------------------------------------------------------------------------
__global__ __launch_bounds__(256) void node_aggregate(
        const float* __restrict__ feat,
        const int*   __restrict__ offsets,
        const int2*  __restrict__ epay,
        float*       __restrict__ agg) {
    const int lane = threadIdx.x & 31;
    const int wave = threadIdx.x >> 5;
    const int node = blockIdx.x * 8 + wave;
    if (node >= NNODES) return;

    const int beg = offsets[node];
    const int end = offsets[node + 1];
    v4f acc = {0.f, 0.f, 0.f, 0.f};

    int i = beg;
    for (; i + 1 < end; i += 2) {           // 2-deep to overlap gather latency
        const int2 p0 = epay[i];
        const int2 p1 = epay[i + 1];
        const v4f f0 = *(const v4f*)(feat + (size_t)p0.x * DIM + lane * 4);
        const v4f f1 = *(const v4f*)(feat + (size_t)p1.x * DIM + lane * 4);
        acc += f0 * __int_as_float(p0.y);
        acc += f1 * __int_as_float(p1.y);
    }
    if (i < end) {
        const int2 p = epay[i];
        const v4f f = *(const v4f*)(feat + (size_t)p.x * DIM + lane * 4);
        acc += f * __int_as_float(p.y);
    }
    *(v4f*)(agg + (size_t)node * DIM + lane * 4) = acc;
}

// ---------------------------------------------------------------------------
// TDM descriptor build + issue: load one 16x128 f32 tile (rows m0..m0+15 of
// io) into LDS at lds_off, with hardware padding of 4 dwords every 128 dwords
// (reproducing the stride-132 conflict-free layout).  ISA 8.3/8.4.
// clang-23 toolchain: 6-arg builtin (g0, g1, g2, g3, g_extra, cpol).
// ---------------------------------------------------------------------------
__device__ __forceinline__ void tdm_load_tile(const float* gsrc, unsigned lds_off) {
    const uint64_t ga = (uint64_t)(uintptr_t)gsrc;
    v4u g0;
    g0.x = 1u;                                   // count=1, user descriptor
    g0.y = lds_off;                              // lds_addr (bytes)
    g0.z = (unsigned)(ga & 0xFFFFFFFFu);         // global_addr[31:0]
    g0.w = (unsigned)((ga >> 32) & 0x1FFFFFFu)   // global_addr[56:32]
         | 0x80000000u;                          // type=2 ("image")
    v8i g1;
    g1[0] = (int)((2u << 16)      // data_size = 4B
                | (1u << 20)      // pad_enable
                | (6u << 22)      // pad_interval: 128 dwords
                | (3u << 25));    // pad_amount : 4 dwords
    g1[1] = (int)(128u << 16);    // tensor_dim0 = 128 (low 16 -> bits 63:48)
    g1[2] = (int)(16u << 16);     // tensor_dim0 hi=0 | tensor_dim1 = 16
    g1[3] = (int)(128u << 16);    // tensor_dim1 hi=0 | tile_dim0 = 128
    g1[4] = 16;                   // tile_dim1 = 16, tile_dim2 = 0
    g1[5] = 128;                  // tensor_dim0_stride = 128
    g1[6] = 0;
    g1[7] = 0;
    v4i z4 = {0, 0, 0, 0};
    v8i z8 = {0, 0, 0, 0, 0, 0, 0, 0};
    __builtin_amdgcn_tensor_load_to_lds(g0, g1, z4, z4, z8, 0);
}

// ---------------------------------------------------------------------------
// GEMM + bias + ReLU, in place on io = d_out: io = relu(io @ W.T + b).
// Block = 8 waves; each block owns MCHUNK consecutive 16-row M-tiles.
// Each wave keeps its 16x128 B-panel (W rows n0..n0+15) in 64 VGPRs and
// loops over the M-tiles, so W is read from L2 only once per block.
// A-tiles are streamed LDS-double-buffered by the Tensor Data Mover:
// wave 0 issues tensor_load_to_lds for tile t+1 while everyone computes
// tile t; completion via s_wait_tensorcnt + workgroup barrier.
// All global reads of a tile precede the barrier before its stores, so the
// in-place update is race-free (distinct row ranges otherwise).
// Layouts per CDNA5 ISA 7.12.2:
//   A 16x4 f32 : lane L holds M = L%16, VGPR r holds K = 2*(L/16) + r
//   B 4x16 f32 : lane L holds N = L%16, VGPR r holds K = 2*(L/16) + r
//   C/D 16x16  : lane L holds N = L%16, VGPR r holds M = r + 8*(L/16)
// ---------------------------------------------------------------------------
__global__ __launch_bounds__(256) void gemm_bias_relu(
        const float* __restrict__ Wm,    // [128(out), 128(in)] row-major
        const float* __restrict__ bias,  // [128]
        float*       __restrict__ io) {  // [N, 128] in/out
    __shared__ float At[2][16 * LDSTR];
    const int tid   = threadIdx.x;
    const int lane  = tid & 31;
    const int wave  = tid >> 5;
    const int col   = lane & 15;
    const int khalf = lane >> 4;
    const int n0    = wave << 4;
    const int mbase = blockIdx.x * (16 * MCHUNK);

    // B-panel (W rows n0..n0+15) resident in registers: 32 x v2f = 64 VGPRs.
    v2f breg[DIM / 4];
    {
        const float* brow = Wm + (size_t)(n0 + col) * DIM + 2 * khalf;
#pragma unroll
        for (int k = 0; k < DIM / 4; ++k)
            breg[k] = *(const v2f*)(brow + 4 * k);
    }
    const float bj = bias[n0 + col];

    // Prime the pipeline: TDM tile 0 -> At[0].
    if (wave == 0) {
        tdm_load_tile(io + (size_t)mbase * DIM, (unsigned)(uintptr_t)&At[0][0]);
        __builtin_amdgcn_s_wait_tensorcnt(0);
    }
    __syncthreads();

    for (int t = 0; t < MCHUNK; ++t) {
        // Prefetch next tile into the alternate buffer (no one reads it yet).
        if (wave == 0 && t + 1 < MCHUNK) {
            tdm_load_tile(io + (size_t)(mbase + (t + 1) * 16) * DIM,
                          (unsigned)(uintptr_t)&At[(t + 1) & 1][0]);
        }

        const float* arow = &At[t & 1][col * LDSTR + 2 * khalf];
        v8f acc = {};
#pragma unroll
        for (int k = 0; k < DIM / 4; ++k) {
            v2f a = *(const v2f*)(arow + 4 * k);
            acc = __builtin_amdgcn_wmma_f32_16x16x4_f32(
                /*neg_a=*/false, a, /*neg_b=*/false, breg[k],
                /*c_mod=*/(short)0, acc, /*reuse_a=*/false, /*reuse_b=*/false);
        }

        const int m0 = mbase + t * 16;
#pragma unroll
        for (int r = 0; r < 8; ++r) {
            const int m = m0 + r + 8 * khalf;
            float v = acc[r] + bj;
            io[(size_t)m * DIM + n0 + col] = v > 0.f ? v : 0.f;
        }

        // Tile t fully consumed; ensure tile t+1 landed before next round.
        if (wave == 0 && t + 1 < MCHUNK) __builtin_amdgcn_s_wait_tensorcnt(0);
        __syncthreads();
    }
}

// ---------------------------------------------------------------------------
extern "C" void kernel_launch(void* const* d_in, const int* in_sizes, int n_in,
                              void* d_out, int out_size, void* d_ws, size_t ws_size,
                              hipStream_t stream) {
    const float*     feature = (const float*)d_in[0];     // [50000,128] f32
    const long long* src     = (const long long*)d_in[1]; // [800000] i64
    const long long* dst     = (const long long*)d_in[2]; // [800000] i64
    const float*     w       = (const float*)d_in[3];     // [800000] f32
    const float*     Wm      = (const float*)d_in[4];     // [128,128] f32
    const float*     bias    = (const float*)d_in[5];     // [128] f32
    float* out = (float*)d_out;                           // [50000,128] f32

    // Workspace layout (~7 MB total)
    int*  offsets = (int*)d_ws;                 // N+1
    int*  cursors = offsets + (NNODES + 1);     // N
    int*  counts  = cursors + NNODES;           // N
    int2* epay    = (int2*)(counts + NNODES);   // E (8B each)

    // 1) zero counts (N divisible by 4)
    zero_v4<<<(NNODES / 4 + 255) / 256, 256, 0, stream>>>((float*)counts, NNODES / 4);

    // 2) histogram dst
    count_edges<<<(NEDGES + 255) / 256, 256, 0, stream>>>(dst, counts);

    // 3) exclusive scan -> offsets & cursors
    scan_offsets<<<1, 1024, 0, stream>>>(counts, offsets, cursors, NNODES);

    // 4) bucket fill (packed src+weight payload)
    fill_edges<<<(NEDGES + 255) / 256, 256, 0, stream>>>(src, dst, w, cursors, epay);

    // 5) per-node gather-reduce into d_out (one wave per node)
    node_aggregate<<<(NNODES + 7) / 8, 256, 0, stream>>>(feature, offsets, epay, out);

    // 6) in-place WMMA GEMM + bias + ReLU, TDM-pipelined, B in registers
    gemm_bias_relu<<<NNODES / (16 * MCHUNK), 256, 0, stream>>>(Wm, bias, out);
}